// DecodeDetections_10874857193790
// MI455X (gfx1250) — compile-verified
//
#include <hip/hip_runtime.h>
#include <math.h>

// Problem constants (from reference): y_pred (32, 24564, 33) f32
#define B_      32
#define N_      24564
#define L_      33
#define C_      20          // L - 13
#define KNMS    400         // NMS_MAX
#define TOPK    200
#define CONF_TH 0.01f
#define IOU_TH  0.45f
#define IMG_WH  512.0f
#define GCAP    1024        // gather capacity for per-(b,c) select
#define FLATN   (C_ * KNMS) // 8000
#define M4      8192        // pow2 pad for final sort

// Monotone float <-> sortable-uint mapping (descending sort by key desc).
__device__ __forceinline__ unsigned f2key(float f) {
    unsigned u = __float_as_uint(f);
    return (u & 0x80000000u) ? ~u : (u | 0x80000000u);
}
__device__ __forceinline__ float key2f(unsigned k) {
    unsigned u = (k & 0x80000000u) ? (k & 0x7FFFFFFFu) : ~k;
    return __uint_as_float(u);
}

// ---------------------------------------------------------------------------
// Kernel 1: decode boxes. Tile rows of y_pred into LDS (coalesced), decode.
// ---------------------------------------------------------------------------
__global__ __launch_bounds__(256) void k_decode(const float* __restrict__ Y,
                                                float4* __restrict__ boxes) {
    __shared__ float tile[256 * L_];
    const long long BN = (long long)B_ * N_;
    const long long a0 = (long long)blockIdx.x * 256;
    const int cnt = (BN - a0 < 256) ? (int)(BN - a0) : 256;
    const int tot = cnt * L_;
    const float* src = Y + a0 * L_;
    for (int t = threadIdx.x; t < tot; t += 256) tile[t] = src[t];
    // Speculative prefetch of next tile -> global_prefetch_b8 (gfx1250 path)
    if (a0 + 256 < BN) __builtin_prefetch(src + 256 * L_ + threadIdx.x * 32, 0, 1);
    __syncthreads();
    const int r = threadIdx.x;
    if (r < cnt) {
        const float* p = &tile[r * L_];
        float cx = p[21] * p[29] * p[27] + p[25];
        float cy = p[22] * p[30] * p[28] + p[26];
        float w  = expf(p[23] * p[31]) * p[27];
        float h  = expf(p[24] * p[32]) * p[28];
        float4 bx;
        bx.x = (cx - 0.5f * w) * IMG_WH;
        bx.y = (cy - 0.5f * h) * IMG_WH;
        bx.z = (cx + 0.5f * w) * IMG_WH;
        bx.w = (cy + 0.5f * h) * IMG_WH;
        boxes[a0 + r] = bx;
    }
}

// ---------------------------------------------------------------------------
// Kernel 2: per-(b,c) exact top-400 by MSB-first radix select (11/11/10 bits)
// then gather key >= pivot and bitonic sort 1024 entries in LDS by
// (key desc, idx asc) — matches jax.lax.top_k tie-breaking.
// Re-reads are L2-resident (104MB input < 192MB L2).
// ---------------------------------------------------------------------------
__global__ __launch_bounds__(256) void k_select(const float* __restrict__ Y,
                                                const float4* __restrict__ boxes,
                                                float4* __restrict__ candBox,
                                                float* __restrict__ candScore) {
    __shared__ unsigned hist[2048];
    __shared__ unsigned gkey[GCAP];
    __shared__ unsigned short gidx[GCAP];
    __shared__ unsigned sh_prefix, sh_need, sh_cnt;

    const int bc = blockIdx.x;
    const int b  = bc / C_;
    const int c  = bc % C_;
    const float* S = Y + (size_t)b * N_ * L_ + (1 + c); // score n at S[n*L_]

    if (threadIdx.x == 0) { sh_prefix = 0u; sh_need = KNMS; }

    const int shifts[3] = {21, 10, 0};
    const int bitsA[3]  = {11, 11, 10};
    for (int p = 0; p < 3; ++p) {
        const int shift = shifts[p];
        const int nb = 1 << bitsA[p];
        for (int t = threadIdx.x; t < nb; t += 256) hist[t] = 0u;
        __syncthreads();
        const unsigned pref = sh_prefix;
        const int hiShift = shift + bitsA[p];
        for (int n = threadIdx.x; n < N_; n += 256) {
            unsigned key = f2key(S[(size_t)n * L_]);
            bool match = (p == 0) || ((key >> hiShift) == pref);
            if (match) atomicAdd(&hist[(key >> shift) & (nb - 1)], 1u);
        }
        __syncthreads();
        if (threadIdx.x == 0) {
            unsigned need = sh_need, acc = 0;
            int d = nb - 1;
            for (; d > 0; --d) {
                unsigned h = hist[d];
                if (acc + h >= need) break;
                acc += h;
            }
            sh_prefix = (pref << bitsA[p]) | (unsigned)d;
            sh_need = need - acc;
        }
        __syncthreads();
    }
    const unsigned pivot = sh_prefix; // exact 32-bit pivot key (rank-400 value)
    if (threadIdx.x == 0) sh_cnt = 0u;
    __syncthreads();

    for (int n = threadIdx.x; n < N_; n += 256) {
        unsigned key = f2key(S[(size_t)n * L_]);
        if (key >= pivot) {
            unsigned pos = atomicAdd(&sh_cnt, 1u);
            if (pos < GCAP) { gkey[pos] = key; gidx[pos] = (unsigned short)n; }
        }
    }
    __syncthreads();
    const int cnt = (int)((sh_cnt < (unsigned)GCAP) ? sh_cnt : (unsigned)GCAP);
    for (int t = cnt + (int)threadIdx.x; t < GCAP; t += 256) {
        gkey[t] = 0u; gidx[t] = 0xFFFFu;
    }
    __syncthreads();

    // Bitonic sort 1024 entries: order = (key desc, idx asc)
    for (unsigned k = 2; k <= GCAP; k <<= 1) {
        for (unsigned j = k >> 1; j > 0; j >>= 1) {
            for (unsigned t = threadIdx.x; t < GCAP / 2; t += 256) {
                unsigned i = 2u * t - (t & (j - 1u));
                unsigned ixj = i + j;
                unsigned ka = gkey[i], kb = gkey[ixj];
                unsigned short ia = gidx[i], ib = gidx[ixj];
                bool aAfterB = (ka < kb) || (ka == kb && ia > ib);
                bool doSwap = ((i & k) == 0u) ? aAfterB : !aAfterB;
                if (doSwap) {
                    gkey[i] = kb; gkey[ixj] = ka;
                    gidx[i] = ib; gidx[ixj] = ia;
                }
            }
            __syncthreads();
        }
    }

    // Emit top-400 scores + gathered boxes
    for (int r = threadIdx.x; r < KNMS; r += 256) {
        unsigned key = gkey[r];
        int anchor = (int)gidx[r];
        candScore[(size_t)bc * KNMS + r] = key2f(key);
        float4 bx = make_float4(0.f, 0.f, 0.f, 0.f);
        if (key != 0u && anchor < N_) bx = boxes[(size_t)b * N_ + anchor];
        candBox[(size_t)bc * KNMS + r] = bx;
    }
}

// ---------------------------------------------------------------------------
// Kernel 3: greedy NMS per (b,c).
//  - Box tile (400 x float4 = 6400B contiguous) staged to LDS with ONE
//    Tensor Data Mover descriptor (tensor_load_to_lds, TENSORcnt path),
//    issued by wave 0 only (TDM ignores EXEC -> one issue per wave otherwise).
//  - Scores staged with per-lane async-to-LDS (ASYNCcnt path).
// Then the exact sequential greedy scan.
// ---------------------------------------------------------------------------
__global__ __launch_bounds__(256) void k_nms(const float4* __restrict__ candBox,
                                             const float* __restrict__ candScore,
                                             float* __restrict__ keptSc) {
    __shared__ float4 sbox[KNMS];
    __shared__ float ssc[KNMS];
    __shared__ float sarea[KNMS];
    __shared__ unsigned ssup[KNMS];

    const int bc = blockIdx.x;
    const float4* gb = candBox + (size_t)bc * KNMS;
    const float* gs = candScore + (size_t)bc * KNMS;

#if defined(__gfx1250__)
    typedef unsigned u32x4 __attribute__((ext_vector_type(4)));
    typedef unsigned u32x8 __attribute__((ext_vector_type(8)));
    if (threadIdx.x < 32) { // wave 0 only: TDM ignores EXEC, issue exactly once
        const unsigned ELEMS = KNMS * 4u; // 1600 dwords, contiguous 1-D tile
        unsigned long long ga = (unsigned long long)(uintptr_t)gb;
        unsigned lds0 = (unsigned)(uintptr_t)&sbox[0];
        u32x4 g0;
        g0[0] = 1u;                                   // count=1 valid descriptor
        g0[1] = lds0;                                 // lds_addr
        g0[2] = (unsigned)(ga & 0xFFFFFFFFu);         // global_addr[31:0]
        g0[3] = (unsigned)((ga >> 32) & 0x01FFFFFFu)  // global_addr[56:32]
              | (2u << 30);                           // type = 2 ("image")
        u32x8 g1;
        g1[0] = (2u << 16);                           // data_size = 4B
        g1[1] = (ELEMS & 0xFFFFu) << 16;              // tensor_dim0[15:0]
        g1[2] = ((ELEMS >> 16) & 0xFFFFu)             // tensor_dim0[31:16]
              | (1u << 16);                           // tensor_dim1[15:0] = 1
        g1[3] = (ELEMS << 16);                        // tile_dim0 = 1600
        g1[4] = 1u;                                   // tile_dim1 = 1
        g1[5] = ELEMS;                                // tensor_dim0_stride
        g1[6] = 0u;
        g1[7] = 0u;
        asm volatile("tensor_load_to_lds %0, %1" :: "s"(g0), "s"(g1) : "memory");
        __builtin_amdgcn_s_wait_tensorcnt(0);
    }
    // Scores: per-lane async global->LDS (ASYNCcnt path)
    for (int j = threadIdx.x; j < KNMS; j += 256) {
        unsigned loffS = (unsigned)(unsigned long long)(uintptr_t)&ssc[j];
        unsigned long long gaS = (unsigned long long)(uintptr_t)(gs + j);
        asm volatile("global_load_async_to_lds_b32 %0, %1, off"
                     :: "v"(loffS), "v"(gaS) : "memory");
    }
    asm volatile("s_wait_asynccnt 0" ::: "memory");
#else
    for (int j = threadIdx.x; j < KNMS; j += 256) { sbox[j] = gb[j]; ssc[j] = gs[j]; }
#endif
    __syncthreads();

    for (int j = threadIdx.x; j < KNMS; j += 256) {
        float4 bx = sbox[j];
        sarea[j] = (bx.z - bx.x) * (bx.w - bx.y);
        ssup[j] = (ssc[j] > CONF_TH) ? 0u : 1u; // suppressed init = !valid
    }
    __syncthreads();

    for (int i = 0; i < KNMS; ++i) {
        bool keep_i = (ssup[i] == 0u);
        if (keep_i) {
            float4 bi = sbox[i];
            float ai = sarea[i];
            for (int j = threadIdx.x; j < KNMS; j += 256) {
                if (j > i) {
                    float4 bj = sbox[j];
                    float ix1 = fmaxf(bi.x, bj.x);
                    float iy1 = fmaxf(bi.y, bj.y);
                    float ix2 = fminf(bi.z, bj.z);
                    float iy2 = fminf(bi.w, bj.w);
                    float inter = fmaxf(ix2 - ix1, 0.f) * fmaxf(iy2 - iy1, 0.f);
                    float iou = inter / (ai + sarea[j] - inter + 1e-8f);
                    if (iou > IOU_TH) ssup[j] = 1u;
                }
            }
        }
        __syncthreads();
    }

    for (int j = threadIdx.x; j < KNMS; j += 256) {
        keptSc[(size_t)bc * KNMS + j] = (ssup[j] == 0u) ? ssc[j] : -1.0f;
    }
}

// ---------------------------------------------------------------------------
// Kernel 4: per-batch top-200 of 8000 via full bitonic sort in LDS (48KB of
// the 320KB WGP budget), then emit [cls+1, score, x1, y1, x2, y2].
// ---------------------------------------------------------------------------
__global__ __launch_bounds__(512) void k_final(const float* __restrict__ keptSc,
                                               const float4* __restrict__ candBox,
                                               float* __restrict__ out) {
    __shared__ unsigned skey[M4];
    __shared__ unsigned short sidx[M4];
    const int b = blockIdx.x;
    const float* fs = keptSc + (size_t)b * FLATN;
    for (int t = threadIdx.x; t < M4; t += 512) {
        if (t < FLATN) { skey[t] = f2key(fs[t]); sidx[t] = (unsigned short)t; }
        else           { skey[t] = 0u;           sidx[t] = 0xFFFFu; }
    }
    __syncthreads();

    for (unsigned k = 2; k <= M4; k <<= 1) {
        for (unsigned j = k >> 1; j > 0; j >>= 1) {
            for (unsigned t = threadIdx.x; t < M4 / 2; t += 512) {
                unsigned i = 2u * t - (t & (j - 1u));
                unsigned ixj = i + j;
                unsigned ka = skey[i], kb = skey[ixj];
                unsigned short ia = sidx[i], ib = sidx[ixj];
                bool aAfterB = (ka < kb) || (ka == kb && ia > ib);
                bool doSwap = ((i & k) == 0u) ? aAfterB : !aAfterB;
                if (doSwap) {
                    skey[i] = kb; skey[ixj] = ka;
                    sidx[i] = ib; sidx[ixj] = ia;
                }
            }
            __syncthreads();
        }
    }

    if (threadIdx.x < TOPK) {
        const int r = threadIdx.x;
        float s = key2f(skey[r]);
        int fi = (int)sidx[r];
        bool valid = (s > 0.0f) && (fi < FLATN);
        float4 bx = make_float4(0.f, 0.f, 0.f, 0.f);
        float cls = 0.0f;
        if (valid) {
            bx = candBox[(size_t)b * FLATN + fi];
            cls = (float)(fi / KNMS);
        }
        float* o = out + ((size_t)b * TOPK + r) * 6;
        o[0] = cls + 1.0f;            // where(valid, cls, 0) + 1
        o[1] = valid ? s : 0.0f;
        o[2] = bx.x; o[3] = bx.y; o[4] = bx.z; o[5] = bx.w;
    }
}

// ---------------------------------------------------------------------------
extern "C" void kernel_launch(void* const* d_in, const int* in_sizes, int n_in,
                              void* d_out, int out_size, void* d_ws, size_t ws_size,
                              hipStream_t stream) {
    (void)in_sizes; (void)n_in; (void)out_size; (void)ws_size;
    const float* Y = (const float*)d_in[0];
    float* out = (float*)d_out;

    // Workspace layout (~18.7 MB total)
    float4* boxes     = (float4*)d_ws;                              // B*N
    float4* candBox   = boxes + (size_t)B_ * N_;                    // B*C*400
    float*  candScore = (float*)(candBox + (size_t)B_ * C_ * KNMS); // B*C*400
    float*  keptSc    = candScore + (size_t)B_ * C_ * KNMS;         // B*C*400

    const int nblk1 = (B_ * N_ + 255) / 256;
    k_decode<<<nblk1, 256, 0, stream>>>(Y, boxes);
    k_select<<<B_ * C_, 256, 0, stream>>>(Y, boxes, candBox, candScore);
    k_nms<<<B_ * C_, 256, 0, stream>>>(candBox, candScore, keptSc);
    k_final<<<B_, 512, 0, stream>>>(keptSc, candBox, out);
}